// CapsuleLayer_46024869544135
// MI455X (gfx1250) — compile-verified
//
#include <hip/hip_runtime.h>
#include <hip/hip_bf16.h>

// Problem constants (match the reference).
#define Bn   32          // batch
#define In   32          // input capsules
#define AIN  16          // input atoms (GEMM K)
#define ODn  10          // output classes
#define OAn  16          // output atoms
#define HWn  144         // 12*12 spatial
#define Kn   160         // OD*OA (GEMM N)
#define NPTS (Bn * HWn)  // 4608 spatial-batch points (GEMM M)

typedef __attribute__((ext_vector_type(2))) float        v2f;
typedef __attribute__((ext_vector_type(8))) float        v8f;
typedef __attribute__((ext_vector_type(4))) unsigned int v4u;
typedef __attribute__((ext_vector_type(8))) int          v8i;
typedef __attribute__((ext_vector_type(4))) int          v4i;

// ---------------------------------------------------------------------------
// Kernel 1: votes[p][i][k] = sum_a x[b,i,a,h,w] * weight[i,a,k]
// One wave per (i, ptile). fp32 WMMA 16x16x4; K=16 = 4 chained WMMAs; A regs
// reused across the 10 column tiles.
// ---------------------------------------------------------------------------
__global__ void __launch_bounds__(256)
caps_votes_wmma(const float* __restrict__ x,
                const float* __restrict__ wgt,
                float* __restrict__ votes) {
    const int lane = threadIdx.x & 31;
    const int half = lane >> 4;        // 0: lanes 0-15, 1: lanes 16-31
    const int row  = lane & 15;        // M for A-loads, N for B-loads/stores

    const int task  = blockIdx.x * 8 + (threadIdx.x >> 5); // 0..9215
    const int ptile = task % (NPTS / 16);                  // 0..287
    const int i     = task / (NPTS / 16);                  // 0..31

    // ---- load A tile: 16 points x 16 atoms of x for this capsule i ----
    const int p = ptile * 16 + row;
    const int b = p / HWn;
    const int s = p - b * HWn;
    const float* xp = x + ((size_t)(b * In + i) * AIN) * HWn + s;

    float a_reg[8];
#pragma unroll
    for (int step = 0; step < 4; ++step) {
        const int a0 = step * 4 + half * 2;          // K base for this lane-half
        a_reg[2 * step]     = xp[(size_t)a0 * HWn];
        a_reg[2 * step + 1] = xp[(size_t)(a0 + 1) * HWn];
    }

    const float* wrow = wgt + (size_t)i * AIN * Kn;  // weight[i, a, k]
#pragma unroll 1
    for (int kt = 0; kt < 10; ++kt) {
        const int kcol = kt * 16 + row;
        v8f c = {};
#pragma unroll
        for (int step = 0; step < 4; ++step) {
            const int a0 = step * 4 + half * 2;
            v2f av, bv;
            av[0] = a_reg[2 * step];
            av[1] = a_reg[2 * step + 1];
            bv[0] = wrow[(size_t)a0 * Kn + kcol];
            bv[1] = wrow[(size_t)(a0 + 1) * Kn + kcol];
            c = __builtin_amdgcn_wmma_f32_16x16x4_f32(
                    false, av, false, bv, (short)0, c, false, false);
        }
#pragma unroll
        for (int r = 0; r < 8; ++r) {
            const int m  = r + half * 8;
            const int pp = ptile * 16 + m;
            votes[((size_t)pp * In + i) * Kn + kt * 16 + row] = c[r];
        }
    }
}

// ---------------------------------------------------------------------------
// Kernel 2: dynamic routing, one 256-thread block per point p = b*144 + s.
// Votes staged into LDS by the Tensor Data Mover (one TENSOR_LOAD_TO_LDS per
// block, issued by wave 0, completion via s_wait_tensorcnt). 3 routing
// iterations run entirely out of LDS.
// ---------------------------------------------------------------------------
__global__ void __launch_bounds__(256)
caps_route(const float* __restrict__ votes,
           const float* __restrict__ bias,
           float* __restrict__ out) {
    __shared__ float sm_votes[In * Kn];   // [i][od*16+oa]  5120 floats = 20 KB
    __shared__ float sm_logits[In * ODn]; // [i][od]         320 floats
    __shared__ float sm_route[In * ODn];  //                 320 floats
    __shared__ float sm_act[Kn];          // [od][oa]        160 floats

    const int tid = threadIdx.x;
    const int p   = blockIdx.x;           // 0..4607
    const int b   = p / HWn;
    const int s   = p - b * HWn;

    // ---- TDM: DMA this point's 5120-float vote block into LDS ----
    if (tid < 32) {                       // wave 0 only (TDM ignores EXEC)
        const unsigned long long ga =
            (unsigned long long)(const void*)votes + (unsigned long long)p * (In * Kn * 4);
        const unsigned int lds_base =
            (unsigned int)(unsigned long long)(const void*)&sm_votes[0];

        // D# group 0: count=1 | lds_addr | global_addr[56:0] | type=2
        v4u g0;
        g0[0] = 1u;                                   // count=1, user mode
        g0[1] = lds_base;                             // lds_addr [63:32]
        g0[2] = (unsigned int)ga;                     // global_addr[31:0]
        g0[3] = (unsigned int)((ga >> 32) & 0x1FFFFFFu) | (2u << 30); // ga[56:32] | type=2

        // D# group 1: 2D tile = 1 row x 5120 elems of 4B, stride 5120
        v8i g1;
        g1[0] = 0x00020000;        // wg_mask=0, data_size=2 (4 bytes)
        g1[1] = 0x14000000;        // tensor_dim0[15:0]=5120 in bits [63:48]
        g1[2] = 0x00010000;        // tensor_dim0[31:16]=0, tensor_dim1 lo16 = 1
        g1[3] = 0x14000000;        // tensor_dim1 hi16 = 0, tile_dim0 = 5120
        g1[4] = 0x00000001;        // tile_dim1 = 1, tile_dim2 = 0
        g1[5] = 5120;              // tensor_dim0_stride[31:0]
        g1[6] = 0;                 // stride0 hi, stride1 lo
        g1[7] = 0;                 // stride1 hi

        v4i gz4 = {};              // groups 2/3: dims >= 2 unused
        v8i gz8 = {};              // extra descriptor words (clang-23 6-arg form)
        __builtin_amdgcn_tensor_load_to_lds(g0, g1, gz4, gz4, gz8, 0);
        __builtin_amdgcn_s_wait_tensorcnt(0);
    }

    for (int idx = tid; idx < In * ODn; idx += 256) sm_logits[idx] = 0.0f;
    __syncthreads();                      // wave0 passed tensorcnt wait -> LDS valid

    const int od = tid >> 4;              // valid when tid < 160
    const int oa = tid & 15;
    (void)oa;

    for (int it = 0; it < 3; ++it) {
        // ---- softmax over od, one thread per input capsule i ----
        if (tid < In) {
            const float* lrow = sm_logits + tid * ODn;
            float mx = lrow[0];
#pragma unroll
            for (int d = 1; d < ODn; ++d) mx = fmaxf(mx, lrow[d]);
            float sum = 0.0f;
            float e[ODn];
#pragma unroll
            for (int d = 0; d < ODn; ++d) { e[d] = __expf(lrow[d] - mx); sum += e[d]; }
            const float inv = 1.0f / sum;
#pragma unroll
            for (int d = 0; d < ODn; ++d) sm_route[tid * ODn + d] = e[d] * inv;
        }
        __syncthreads();

        // ---- preact[od][oa] = sum_i route[i][od]*votes[i][od][oa] + bias ----
        float acc = 0.0f;
        if (tid < Kn) {
            acc = bias[tid];
#pragma unroll
            for (int i = 0; i < In; ++i)
                acc = fmaf(sm_route[i * ODn + od], sm_votes[i * Kn + tid], acc);
        }

        // ---- squash over oa: shfl groups of 16 (aligned within wave32) ----
        float ss = acc * acc;
#pragma unroll
        for (int off = 8; off >= 1; off >>= 1) ss += __shfl_xor(ss, off, 16);
        const float act = acc * __fsqrt_rn(ss) / (1.0f + ss); // v*|v|/(1+|v|^2)
        if (tid < Kn) sm_act[tid] = act;
        __syncthreads();

        if (it < 2) {
            // ---- logits[i][od] += sum_oa votes[i][od][oa]*act[od][oa] ----
            for (int idx = tid; idx < In * ODn; idx += 256) {
                const int ii = idx / ODn;
                const int dd = idx - ii * ODn;
                float dot = 0.0f;
#pragma unroll
                for (int a = 0; a < OAn; ++a)
                    dot = fmaf(sm_votes[ii * Kn + dd * 16 + a], sm_act[dd * 16 + a], dot);
                sm_logits[idx] += dot;
            }
            __syncthreads();
        } else {
            if (tid < Kn) out[((size_t)b * Kn + tid) * HWn + s] = act;
        }
    }
}

// ---------------------------------------------------------------------------
// Launch. Workspace holds votes [p][i][k]: 4608*32*160*4 = 94,371,840 bytes.
// ---------------------------------------------------------------------------
extern "C" void kernel_launch(void* const* d_in, const int* in_sizes, int n_in,
                              void* d_out, int out_size, void* d_ws, size_t ws_size,
                              hipStream_t stream) {
    const float* x    = (const float*)d_in[0];
    const float* wgt  = (const float*)d_in[1];
    const float* bias = (const float*)d_in[2];
    float* out   = (float*)d_out;
    float* votes = (float*)d_ws;

    caps_votes_wmma<<<(In * (NPTS / 16)) / 8, 256, 0, stream>>>(x, wgt, votes);
    caps_route<<<NPTS, 256, 0, stream>>>(votes, bias, out);
}